// ForwardWarp_weight_84619445666296
// MI455X (gfx1250) — compile-verified
//
#include <hip/hip_runtime.h>

// Forward warp (Gaussian splatting) for MI455X / gfx1250.
//
// Shapes fixed by the reference: img/counts (8,1,720,1280) f32, flo (8,2,720,1280) f32.
// Output: concat( imgw/(o+EPS) , o ), each (8,1,720,1280) f32.
//
// Strategy: pure scatter-add with hardware fp32 global atomics (L2-resident
// accumulator: 59 MB << 192 MB L2), non-temporal loads for the streamed
// inputs, vectorized zero-init + finalize. No WMMA: this workload has no
// matrix product; its ceiling is L2 atomic + HBM streaming bandwidth.
//
// NOTE: fw_splat_kernel is deliberately the first function in this file so
// the compile-loop's assembly snippet shows its body (atomic + exp path).

#define N_ 8
#define H_ 720
#define W_ 1280
#define HW_ (H_ * W_)          // 921600
#define NHW_ (N_ * HW_)        // 7372800
#define EPS_ 1e-6f

// Native vector type: required for __builtin_nontemporal_{load,store}
// (HIP's float4 is a class and is rejected by the builtin).
typedef float v4f __attribute__((ext_vector_type(4)));

__global__ __launch_bounds__(256) void fw_splat_kernel(const float* __restrict__ img,
                                                       const float* __restrict__ counts,
                                                       const float* __restrict__ flo,
                                                       float* __restrict__ acc_img,
                                                       float* __restrict__ acc_one) {
    int idx = blockIdx.x * blockDim.x + threadIdx.x;
    if (idx >= NHW_) return;

    int n   = idx / HW_;
    int rem = idx - n * HW_;
    int r   = rem / W_;
    int c   = rem - r * W_;

    // Streaming, read-once inputs: non-temporal so they don't displace the
    // L2-resident accumulator lines.
    float iv = __builtin_nontemporal_load(img + idx);
    float cv = __builtin_nontemporal_load(counts + idx);
    const float* fb = flo + (size_t)n * 2 * HW_;
    float y = __builtin_nontemporal_load(fb + rem);        // channel 0: shift along W
    float x = __builtin_nontemporal_load(fb + HW_ + rem);  // channel 1: shift along H

    float x1f = floorf(x);
    float y1f = floorf(y);
    float fx = x - x1f;          // in [0,1)
    float fy = y - y1f;
    float gx = 1.f - fx;
    float gy = 1.f - fy;
    float dx1 = fx * fx, dx2 = gx * gx;
    float dy1 = fy * fy, dy2 = gy * gy;

    float w11 = __expf(-(dx1 + dy1));   // corner (x1, y1)
    float w12 = __expf(-(dx1 + dy2));   // corner (x1, y2)
    float w21 = __expf(-(dx2 + dy1));   // corner (x2, y1)
    float w22 = __expf(-(dx2 + dy2));   // corner (x2, y2)
    float s = w11 + w12 + w21 + w22;
    float cw = cv / s;                  // counts * (1/sum): fold normalization in once
    w11 *= cw; w12 *= cw; w21 *= cw; w22 *= cw;

    int ix1 = r + (int)x1f;    // floor values are integer-exact in f32 here
    int iy1 = c + (int)y1f;
    int ix2 = ix1 + 1;
    int iy2 = iy1 + 1;
    const int base = n * HW_;

    // 4 corners x 2 accumulators: fire-and-forget global_atomic_add_f32.
    // unsafeAtomicAdd guarantees the native FP atomic (no CAS loop).
    #define FW_EMIT(tr, tc, w)                                            \
        if ((unsigned)(tr) < (unsigned)H_ && (unsigned)(tc) < (unsigned)W_) { \
            int o_ = base + (tr) * W_ + (tc);                             \
            unsafeAtomicAdd(acc_img + o_, iv * (w));                      \
            unsafeAtomicAdd(acc_one + o_, (w));                           \
        }
    FW_EMIT(ix1, iy1, w11)
    FW_EMIT(ix1, iy2, w12)
    FW_EMIT(ix2, iy1, w21)
    FW_EMIT(ix2, iy2, w22)
    #undef FW_EMIT
}

__global__ __launch_bounds__(256) void fw_zero_kernel(v4f* __restrict__ p, int n4) {
    int i = blockIdx.x * blockDim.x + threadIdx.x;
    if (i < n4) p[i] = (v4f){0.f, 0.f, 0.f, 0.f};
}

__global__ __launch_bounds__(256) void fw_finalize_kernel(v4f* __restrict__ out0,
                                                          const v4f* __restrict__ out1,
                                                          int n4) {
    int i = blockIdx.x * blockDim.x + threadIdx.x;
    if (i >= n4) return;
    // Both halves are read exactly once here and never again -> NT loads.
    v4f a = __builtin_nontemporal_load(out0 + i);
    v4f b = __builtin_nontemporal_load(out1 + i);
    v4f r;
    r.x = a.x / (b.x + EPS_);
    r.y = a.y / (b.y + EPS_);
    r.z = a.z / (b.z + EPS_);
    r.w = a.w / (b.w + EPS_);
    __builtin_nontemporal_store(r, out0 + i);  // final result, never re-read
}

extern "C" void kernel_launch(void* const* d_in, const int* in_sizes, int n_in,
                              void* d_out, int out_size, void* d_ws, size_t ws_size,
                              hipStream_t stream) {
    const float* img    = (const float*)d_in[0];
    const float* counts = (const float*)d_in[1];
    const float* flo    = (const float*)d_in[2];

    float* out     = (float*)d_out;
    float* acc_img = out;          // first half:  imgw accumulator -> final imgw/(o+eps)
    float* acc_one = out + NHW_;   // second half: o accumulator (final output as-is)

    // 1) Zero both accumulator halves (also warms L2 for the atomics).
    int n4_all = (2 * NHW_) / 4;                        // 3,686,400
    fw_zero_kernel<<<n4_all / 256, 256, 0, stream>>>((v4f*)out, n4_all);

    // 2) Scatter-add splat: one thread per source pixel, coalesced reads.
    fw_splat_kernel<<<NHW_ / 256, 256, 0, stream>>>(img, counts, flo, acc_img, acc_one);

    // 3) Pointwise finalize of the first half in place.
    int n4_half = NHW_ / 4;                             // 1,843,200
    fw_finalize_kernel<<<n4_half / 256, 256, 0, stream>>>((v4f*)acc_img,
                                                          (const v4f*)acc_one, n4_half);
}